// SAGE_4191888081322
// MI455X (gfx1250) — compile-verified
//
#include <hip/hip_runtime.h>
#include <hip/hip_bf16.h>

typedef __attribute__((ext_vector_type(2))) float v2f;
typedef __attribute__((ext_vector_type(8))) float v8f;

#define NN 50000
#define NE 800000
#define CH 128          // IN_CH == HID == 128
#define OUTC 64
#define LDSS 132        // padded LDS row stride (128 % 64 == 0 would 16-way conflict)

// ---------------- degree / inverse-degree ----------------
__global__ void sage_deg_kernel(const long long* __restrict__ dst,
                                float* __restrict__ cnt) {
    int e = blockIdx.x * blockDim.x + threadIdx.x;
    if (e < NE) atomicAdd(cnt + (int)dst[e], 1.0f);
}

__global__ void sage_inv_kernel(const float* __restrict__ cnt,
                                float* __restrict__ inv) {
    int i = blockIdx.x * blockDim.x + threadIdx.x;
    if (i < NN) inv[i] = 1.0f / fmaxf(cnt[i], 1.0f);
}

// ---------------- gather + scatter-add (one wave per edge) ----------------
__global__ void sage_scatter_kernel(const float* __restrict__ h,
                                    const long long* __restrict__ src,
                                    const long long* __restrict__ dst,
                                    float* __restrict__ agg) {
    int wave = (blockIdx.x * blockDim.x + threadIdx.x) >> 5;
    int lane = threadIdx.x & 31;
    if (wave >= NE) return;
    int s = (int)src[wave];
    int d = (int)dst[wave];
    float4 v = *(const float4*)(h + (size_t)s * CH + lane * 4);
    float* p = agg + (size_t)d * CH + lane * 4;
    atomicAdd(p + 0, v.x);
    atomicAdd(p + 1, v.y);
    atomicAdd(p + 2, v.z);
    atomicAdd(p + 3, v.w);
}

// ---------------- fused SAGE GEMM: out = relu(mean@Wl + hin@Wr + b) ------
// Block handles 64 rows x NOUT cols. One wave per 16-col N-tile, 4 M-subtiles.
// f32 WMMA 16x16x4: A frag = v2f (K pair, lane half selects K+0/1 vs K+2/3),
// B frag = v2f, C/D = v8f.
template <int NOUT, bool HAS_AGG, bool RELU>
__global__ void __launch_bounds__(32 * (NOUT / 16))
sage_gemm_kernel(const float* __restrict__ hin,   // NN x CH
                 const float* __restrict__ agg,   // NN x CH (HAS_AGG)
                 const float* __restrict__ inv,   // NN      (HAS_AGG)
                 const float* __restrict__ Wl,    // CH x NOUT (HAS_AGG)
                 const float* __restrict__ Wr,    // CH x NOUT
                 const float* __restrict__ bias,  // NOUT
                 float* __restrict__ out) {       // NN x NOUT
    __shared__ float sX[64 * LDSS];
    __shared__ float sM[HAS_AGG ? 64 * LDSS : 4];

    const int tid    = threadIdx.x;
    const int nthr   = 32 * (NOUT / 16);
    const int mbase  = blockIdx.x * 64;

    // Stage 64x128 tile(s) into LDS as float4; fold mean = agg * inv_cnt.
    for (int i = tid; i < 64 * 32; i += nthr) {
        int row = i >> 5;
        int c4  = (i & 31) * 4;
        int gr  = mbase + row;
        float4 vx = make_float4(0.f, 0.f, 0.f, 0.f);
        if (gr < NN) vx = *(const float4*)(hin + (size_t)gr * CH + c4);
        *(float4*)(&sX[row * LDSS + c4]) = vx;
        if constexpr (HAS_AGG) {
            float4 va = make_float4(0.f, 0.f, 0.f, 0.f);
            if (gr < NN) {
                float s = inv[gr];
                va = *(const float4*)(agg + (size_t)gr * CH + c4);
                va.x *= s; va.y *= s; va.z *= s; va.w *= s;
            }
            *(float4*)(&sM[row * LDSS + c4]) = va;
        }
    }
    __syncthreads();

    const int wv    = tid >> 5;          // N-tile index
    const int lane  = tid & 31;
    const int n     = wv * 16 + (lane & 15);
    const int khalf = (lane >> 4) * 2;   // low lanes K+0/1, high lanes K+2/3

    v8f acc[4];
#pragma unroll
    for (int m = 0; m < 4; ++m) acc[m] = (v8f){0,0,0,0,0,0,0,0};

#pragma unroll 4
    for (int k = 0; k < CH; k += 4) {
        v2f br; // B frag of Wr (dense/root path)
        br.x = Wr[(size_t)(k + khalf) * NOUT + n];
        br.y = Wr[(size_t)(k + khalf + 1) * NOUT + n];
        v2f bl2;
        if constexpr (HAS_AGG) {
            bl2.x = Wl[(size_t)(k + khalf) * NOUT + n];
            bl2.y = Wl[(size_t)(k + khalf + 1) * NOUT + n];
        }
#pragma unroll
        for (int m = 0; m < 4; ++m) {
            int r = m * 16 + (lane & 15);
            v2f ax = *(const v2f*)&sX[r * LDSS + k + khalf];
            acc[m] = __builtin_amdgcn_wmma_f32_16x16x4_f32(
                false, ax, false, br, (short)0, acc[m], false, false);
            if constexpr (HAS_AGG) {
                v2f am = *(const v2f*)&sM[r * LDSS + k + khalf];
                acc[m] = __builtin_amdgcn_wmma_f32_16x16x4_f32(
                    false, am, false, bl2, (short)0, acc[m], false, false);
            }
        }
    }

    const float b = bias[n];
#pragma unroll
    for (int m = 0; m < 4; ++m) {
        int rbase = mbase + m * 16;
#pragma unroll
        for (int v = 0; v < 8; ++v) {
            int row = rbase + ((lane < 16) ? v : v + 8);
            if (row < NN) {
                float val = acc[m][v] + b;
                if constexpr (RELU) val = fmaxf(val, 0.0f);
                out[(size_t)row * NOUT + n] = val;
            }
        }
    }
}

// ---------------- log_softmax over 64 cols (one wave per row) ------------
__global__ void sage_lsm_kernel(const float* __restrict__ logits,
                                float* __restrict__ out) {
    int row  = (blockIdx.x * blockDim.x + threadIdx.x) >> 5;
    int lane = threadIdx.x & 31;
    if (row >= NN) return;
    const float* p = logits + (size_t)row * OUTC;
    float a = p[lane];
    float b = p[lane + 32];
    float mx = fmaxf(a, b);
#pragma unroll
    for (int o = 16; o; o >>= 1) mx = fmaxf(mx, __shfl_xor(mx, o, 32));
    float s = expf(a - mx) + expf(b - mx);
#pragma unroll
    for (int o = 16; o; o >>= 1) s += __shfl_xor(s, o, 32);
    float lse = mx + logf(s);
    float* q = out + (size_t)row * OUTC;
    q[lane]      = a - lse;
    q[lane + 32] = b - lse;
}

// ------------------------------ launcher ---------------------------------
extern "C" void kernel_launch(void* const* d_in, const int* in_sizes, int n_in,
                              void* d_out, int out_size, void* d_ws, size_t ws_size,
                              hipStream_t stream) {
    const float*     x    = (const float*)d_in[0];
    const long long* ei   = (const long long*)d_in[1]; // int64 per reference
    const float*     Wl0  = (const float*)d_in[2];
    const float*     bl0  = (const float*)d_in[3];
    const float*     Wr0  = (const float*)d_in[4];
    const float*     Wl1  = (const float*)d_in[5];
    const float*     bl1  = (const float*)d_in[6];
    const float*     Wr1  = (const float*)d_in[7];
    const float*     Wl2  = (const float*)d_in[8];
    const float*     bl2  = (const float*)d_in[9];
    const float*     Wr2  = (const float*)d_in[10];
    const float*     Wlin = (const float*)d_in[11];
    const float*     blin = (const float*)d_in[12];
    const long long* src  = ei;
    const long long* dst  = ei + NE;

    char* ws = (char*)d_ws;
    const size_t featB = (size_t)NN * CH * sizeof(float);   // 25.6 MB
    float* hA   = (float*)(ws);                              // layer outputs (ping)
    float* hB   = (float*)(ws + featB);                      // layer outputs (pong)
    float* agg  = (float*)(ws + 2 * featB);                  // neighbor sums
    float* cnt  = (float*)(ws + 3 * featB);                  // degrees
    float* inv  = (float*)(ws + 3 * featB + ((size_t)NN * 4 + 255 & ~(size_t)255));

    const int edgeWaveBlocks = (NE * 32 + 255) / 256;        // 1 wave / edge
    const int gemmBlocks     = (NN + 63) / 64;

    // degrees (shared by all layers)
    hipMemsetAsync(cnt, 0, (size_t)NN * 4, stream);
    sage_deg_kernel<<<(NE + 255) / 256, 256, 0, stream>>>(dst, cnt);
    sage_inv_kernel<<<(NN + 255) / 256, 256, 0, stream>>>(cnt, inv);

    // layer 0: x -> hA
    hipMemsetAsync(agg, 0, featB, stream);
    sage_scatter_kernel<<<edgeWaveBlocks, 256, 0, stream>>>(x, src, dst, agg);
    sage_gemm_kernel<CH, true, true><<<gemmBlocks, 256, 0, stream>>>(
        x, agg, inv, Wl0, Wr0, bl0, hA);

    // layer 1: hA -> hB
    hipMemsetAsync(agg, 0, featB, stream);
    sage_scatter_kernel<<<edgeWaveBlocks, 256, 0, stream>>>(hA, src, dst, agg);
    sage_gemm_kernel<CH, true, true><<<gemmBlocks, 256, 0, stream>>>(
        hA, agg, inv, Wl1, Wr1, bl1, hB);

    // layer 2: hB -> hA
    hipMemsetAsync(agg, 0, featB, stream);
    sage_scatter_kernel<<<edgeWaveBlocks, 256, 0, stream>>>(hB, src, dst, agg);
    sage_gemm_kernel<CH, true, true><<<gemmBlocks, 256, 0, stream>>>(
        hB, agg, inv, Wl2, Wr2, bl2, hA);

    // head: logits = hA @ Wlin + blin  (reuse hB for logits, only NN*64 floats)
    sage_gemm_kernel<OUTC, false, false><<<gemmBlocks, 128, 0, stream>>>(
        hA, nullptr, nullptr, nullptr, Wlin, blin, hB);

    // log_softmax -> d_out
    sage_lsm_kernel<<<(NN * 32 + 255) / 256, 256, 0, stream>>>(hB, (float*)d_out);
}